// Memory_55954833932314
// MI455X (gfx1250) — compile-verified
//
#include <hip/hip_runtime.h>
#include <math.h>

#define B_   128
#define M_   512
#define W_   64
#define R_   4
#define IN_  512
#define OUTP 471
#define OPAD 480
#define EPS_   1e-6f
#define DELTA_ 1e-6f

typedef __attribute__((ext_vector_type(2))) float v2f;
typedef __attribute__((ext_vector_type(8))) float v8f;

__device__ __forceinline__ float sigm_(float x) { return 1.f / (1.f + expf(-x)); }
__device__ __forceinline__ float softplus_(float x) {
  return fmaxf(x, 0.f) + log1pf(expf(-fabsf(x)));
}

// ---------------------------------------------------------------------------
// Kernel 1: concatenate the 10 projection weight matrices into wcat[480][512]
// (rows 471..479 zero-padded) and biases into bias[480].
// Segment offsets: rk 0..255, rs 256..259, wk 260..323, ws 324, ev 325..388,
//                  wv 389..452, fg 453..456, ag 457, wg 458, rm 459..470
// ---------------------------------------------------------------------------
__global__ void pack_w_kernel(
    const float* rk_w, const float* rs_w, const float* wk_w, const float* ws_w,
    const float* ev_w, const float* wv_w, const float* fg_w, const float* ag_w,
    const float* wg_w, const float* rm_w,
    const float* rk_b, const float* rs_b, const float* wk_b, const float* ws_b,
    const float* ev_b, const float* wv_b, const float* fg_b, const float* ag_b,
    const float* wg_b, const float* rm_b,
    float* __restrict__ wcat, float* __restrict__ bias)
{
  int i = blockIdx.x * blockDim.x + threadIdx.x;
  if (i >= OPAD * IN_) return;
  int o = i >> 9, k = i & (IN_ - 1);
  const float* w = nullptr; const float* bb = nullptr; int row = 0;
  if      (o < 256) { w = rk_w; bb = rk_b; row = o;       }
  else if (o < 260) { w = rs_w; bb = rs_b; row = o - 256; }
  else if (o < 324) { w = wk_w; bb = wk_b; row = o - 260; }
  else if (o < 325) { w = ws_w; bb = ws_b; row = o - 324; }
  else if (o < 389) { w = ev_w; bb = ev_b; row = o - 325; }
  else if (o < 453) { w = wv_w; bb = wv_b; row = o - 389; }
  else if (o < 457) { w = fg_w; bb = fg_b; row = o - 453; }
  else if (o < 458) { w = ag_w; bb = ag_b; row = o - 457; }
  else if (o < 459) { w = wg_w; bb = wg_b; row = o - 458; }
  else if (o < 471) { w = rm_w; bb = rm_b; row = o - 459; }
  wcat[i] = w ? w[row * IN_ + k] : 0.f;
  if (k == 0) bias[o] = bb ? bb[row] : 0.f;
}

// ---------------------------------------------------------------------------
// Kernel 2: proj[128][480] = x[128][512] @ wcat[480][512]^T  via f32 WMMA.
// One wave per 16x16 tile. A 16x4 frag: lane m (0-15) holds K={0,1} (VGPR0/1),
// lane m+16 holds K={2,3}. B 4x16 frag mirrors A (lane = N column of W^T,
// i.e. a row of wcat). C/D: VGPR j -> M=j (lanes 0-15) / M=j+8 (lanes 16-31).
// ---------------------------------------------------------------------------
__global__ __launch_bounds__(32) void proj_gemm_wmma(
    const float* __restrict__ x, const float* __restrict__ wcat,
    float* __restrict__ proj)
{
  int lane = threadIdx.x;
  int mt = blockIdx.x;          // 0..7   (128 rows)
  int nt = blockIdx.y;          // 0..29  (480 cols)
  int rsel = lane & 15;
  int kh   = (lane >> 4) << 1;  // 0 or 2
  const float* pa = x    + (size_t)(mt * 16 + rsel) * IN_ + kh;
  const float* pb = wcat + (size_t)(nt * 16 + rsel) * IN_ + kh;

  v8f acc;
  #pragma unroll
  for (int j = 0; j < 8; ++j) acc[j] = 0.f;

  #pragma unroll 8
  for (int k0 = 0; k0 < IN_; k0 += 4) {
    v2f a, b;
    a[0] = pa[k0]; a[1] = pa[k0 + 1];
    b[0] = pb[k0]; b[1] = pb[k0 + 1];
    acc = __builtin_amdgcn_wmma_f32_16x16x4_f32(
        false, a, false, b, (short)0, acc, false, false);
  }

  int mrow = mt * 16 + (lane >> 4) * 8;
  int ncol = nt * 16 + rsel;
  #pragma unroll
  for (int j = 0; j < 8; ++j)
    proj[(size_t)(mrow + j) * OPAD + ncol] = acc[j];
}

// ---------------------------------------------------------------------------
// Kernel 3: activations + normalized keys + u2 (retention * usage update).
// One block (256 threads) per batch element.
// ---------------------------------------------------------------------------
__global__ __launch_bounds__(256) void act_kernel(
    const float* __restrict__ proj, const float* __restrict__ bias,
    const float* __restrict__ usage, const float* __restrict__ wwts,
    const float* __restrict__ rdw,
    float* __restrict__ kn, float* __restrict__ wkn, float* __restrict__ rs,
    float* __restrict__ wstr, float* __restrict__ er, float* __restrict__ wv,
    float* __restrict__ ag, float* __restrict__ wg, float* __restrict__ rm,
    float* __restrict__ u2)
{
  __shared__ float s_act[OPAD];
  __shared__ float s_norm[5];
  int b = blockIdx.x, tid = threadIdx.x;

  for (int o = tid; o < OUTP; o += 256) s_act[o] = proj[b * OPAD + o] + bias[o];
  __syncthreads();

  for (int o = tid; o < OUTP; o += 256) {
    float v = s_act[o];
    if      (o < 256) v = tanhf(v);       // read keys
    else if (o < 260) v = softplus_(v);   // read strengths
    else if (o < 324) v = tanhf(v);       // write key
    else if (o < 325) v = softplus_(v);   // write strength
    else if (o < 389) v = sigm_(v);       // erase vector
    else if (o < 453) v = tanhf(v);       // write vector
    else if (o < 459) v = sigm_(v);       // fg, ag, wg
    // 459..470 (read modes) stay raw; softmax below
    s_act[o] = v;
  }
  __syncthreads();

  if (tid < 5) {
    int base = (tid < 4) ? tid * 64 : 260;
    float s = 0.f;
    for (int w2 = 0; w2 < 64; ++w2) { float v = s_act[base + w2]; s += v * v; }
    s_norm[tid] = sqrtf(s);
  }
  __syncthreads();

  if (tid < 256) kn[b * 256 + tid] = s_act[tid] / (s_norm[tid >> 6] + EPS_);
  if (tid < 64) {
    wkn[b * 64 + tid] = s_act[260 + tid] / (s_norm[4] + EPS_);
    er [b * 64 + tid] = s_act[325 + tid];
    wv [b * 64 + tid] = s_act[389 + tid];
  }
  if (tid < 4) {
    rs[b * 4 + tid] = s_act[256 + tid];
    float a0 = s_act[459 + tid * 3], a1 = s_act[460 + tid * 3], a2 = s_act[461 + tid * 3];
    float mx = fmaxf(a0, fmaxf(a1, a2));
    float e0 = expf(a0 - mx), e1 = expf(a1 - mx), e2 = expf(a2 - mx);
    float inv = 1.f / (e0 + e1 + e2);
    rm[(b * 4 + tid) * 3 + 0] = e0 * inv;
    rm[(b * 4 + tid) * 3 + 1] = e1 * inv;
    rm[(b * 4 + tid) * 3 + 2] = e2 * inv;
  }
  if (tid == 0) { wstr[b] = s_act[324]; ag[b] = s_act[457]; wg[b] = s_act[458]; }

  for (int m = tid; m < M_; m += 256) {
    float us = usage[b * M_ + m];
    float u  = us + (1.f - us) * wwts[b * M_ + m];   // prod over singleton axis
    float psi = 1.f;
    #pragma unroll
    for (int r = 0; r < R_; ++r)
      psi *= (1.f - s_act[453 + r] * rdw[(b * R_ + r) * M_ + m]);
    u *= psi;
    u2[b * M_ + m] = DELTA_ + (1.f - DELTA_) * u;
  }
}

// ---------------------------------------------------------------------------
// Kernel 4: write content weights = softmax_m(cos(memory, wkey)*wstrength).
// One block (512 threads = one per m) per batch element.
// ---------------------------------------------------------------------------
__global__ __launch_bounds__(512) void wcw_kernel(
    const float* __restrict__ memory, const float* __restrict__ wkn,
    const float* __restrict__ wstr, float* __restrict__ wcw)
{
  __shared__ float s_k[64];
  __shared__ float red[512];
  int b = blockIdx.x, m = threadIdx.x;
  if (m < 64) s_k[m] = wkn[b * 64 + m];
  __syncthreads();

  const float* row = memory + ((size_t)b * M_ + m) * W_;
  float ss = 0.f, dot = 0.f;
  #pragma unroll 8
  for (int w = 0; w < W_; ++w) { float v = row[w]; ss += v * v; dot += v * s_k[w]; }
  float logit = (dot / (sqrtf(ss) + EPS_)) * wstr[b];

  red[m] = logit; __syncthreads();
  for (int s = 256; s > 0; s >>= 1) { if (m < s) red[m] = fmaxf(red[m], red[m + s]); __syncthreads(); }
  float mx = red[0]; __syncthreads();
  float e = expf(logit - mx);
  red[m] = e; __syncthreads();
  for (int s = 256; s > 0; s >>= 1) { if (m < s) red[m] += red[m + s]; __syncthreads(); }
  wcw[b * M_ + m] = e / red[0];
}

// ---------------------------------------------------------------------------
// Kernel 5: allocation weighting. Per batch: stable ascending bitonic sort of
// (u2, index), exclusive product scan, scatter back, then write weights ww.
// ---------------------------------------------------------------------------
__global__ __launch_bounds__(256) void alloc_kernel(
    const float* __restrict__ u2, const float* __restrict__ wcw,
    const float* __restrict__ ag, const float* __restrict__ wg,
    float* __restrict__ ww)
{
  __shared__ float sv[512];
  __shared__ int   si[512];
  __shared__ float sa[512], sb[512], s_alloc[512];
  int b = blockIdx.x, tid = threadIdx.x;

  for (int t = tid; t < 512; t += 256) { sv[t] = u2[b * M_ + t]; si[t] = t; }
  __syncthreads();

  // bitonic sort, ascending, index tiebreak (matches stable jnp.argsort)
  for (int k = 2; k <= 512; k <<= 1) {
    for (int j = k >> 1; j > 0; j >>= 1) {
      int p = tid;
      int i  = ((p & ~(j - 1)) << 1) | (p & (j - 1));
      int x2 = i | j;
      float v0 = sv[i], v1 = sv[x2];
      int   a0 = si[i], a1 = si[x2];
      bool up = ((i & k) == 0);
      bool gt = (v0 > v1) || (v0 == v1 && a0 > a1);
      if (gt == up) { sv[i] = v1; sv[x2] = v0; si[i] = a1; si[x2] = a0; }
      __syncthreads();
    }
  }

  // inclusive product scan (Hillis-Steele, ping-pong)
  for (int t = tid; t < 512; t += 256) sa[t] = sv[t];
  __syncthreads();
  float* src = sa; float* dst = sb;
  for (int d = 1; d < 512; d <<= 1) {
    for (int t = tid; t < 512; t += 256) dst[t] = (t >= d) ? src[t] * src[t - d] : src[t];
    __syncthreads();
    float* tmp = src; src = dst; dst = tmp;
  }

  for (int t = tid; t < 512; t += 256) {
    float excl = (t == 0) ? 1.f : src[t - 1];
    s_alloc[si[t]] = (1.f - sv[t]) * excl;
  }
  __syncthreads();

  float agv = ag[b], wgv = wg[b];
  for (int t = tid; t < 512; t += 256)
    ww[b * M_ + t] = wgv * (agv * s_alloc[t] + (1.f - agv) * wcw[b * M_ + t]);
}

// ---------------------------------------------------------------------------
// Kernel 6: memory update: mem = memory*(1 - ww*erase) + ww*write_vec
// ---------------------------------------------------------------------------
__global__ __launch_bounds__(256) void memupd_kernel(
    const float* __restrict__ memory, const float* __restrict__ ww,
    const float* __restrict__ er, const float* __restrict__ wv,
    float* __restrict__ mem)
{
  int i = blockIdx.x * blockDim.x + threadIdx.x;
  int w  = i & (W_ - 1);
  int bm = i >> 6;
  int b  = bm >> 9;
  float wwv = ww[bm];
  mem[i] = memory[i] * (1.f - wwv * er[b * W_ + w]) + wwv * wv[b * W_ + w];
}

// ---------------------------------------------------------------------------
// Kernel 7: read content weights = softmax_m(cos(mem, read_key_r)*strength_r)
// ---------------------------------------------------------------------------
__global__ __launch_bounds__(512) void cw_kernel(
    const float* __restrict__ mem, const float* __restrict__ kn,
    const float* __restrict__ rs, float* __restrict__ cw)
{
  __shared__ float s_kn[256];
  __shared__ float red[512];
  int b = blockIdx.x, m = threadIdx.x;
  if (m < 256) s_kn[m] = kn[b * 256 + m];
  __syncthreads();

  const float* row = mem + ((size_t)b * M_ + m) * W_;
  float ss = 0.f, d0 = 0.f, d1 = 0.f, d2 = 0.f, d3 = 0.f;
  #pragma unroll 8
  for (int w = 0; w < W_; ++w) {
    float v = row[w];
    ss += v * v;
    d0 += v * s_kn[w];
    d1 += v * s_kn[64 + w];
    d2 += v * s_kn[128 + w];
    d3 += v * s_kn[192 + w];
  }
  float inv = 1.f / (sqrtf(ss) + EPS_);
  float lg[4];
  lg[0] = d0 * inv * rs[b * 4 + 0];
  lg[1] = d1 * inv * rs[b * 4 + 1];
  lg[2] = d2 * inv * rs[b * 4 + 2];
  lg[3] = d3 * inv * rs[b * 4 + 3];

  for (int r = 0; r < R_; ++r) {
    float l = lg[r];
    red[m] = l; __syncthreads();
    for (int s = 256; s > 0; s >>= 1) { if (m < s) red[m] = fmaxf(red[m], red[m + s]); __syncthreads(); }
    float mx = red[0]; __syncthreads();
    float e = expf(l - mx);
    red[m] = e; __syncthreads();
    for (int s = 256; s > 0; s >>= 1) { if (m < s) red[m] += red[m + s]; __syncthreads(); }
    float sum = red[0]; __syncthreads();
    cw[(b * R_ + r) * M_ + m] = e / sum;
  }
}

// ---------------------------------------------------------------------------
// Kernel 8: fwd[b,r,i] = sum_{j!=i} L[i,j]*rw[r,j], L fused on the fly
// (never materialize the 134MB link matrix result).
// One wave per row i; grid (B, 64 row-groups), 8 waves/block.
// ---------------------------------------------------------------------------
__global__ __launch_bounds__(256) void link_fwd_kernel(
    const float* __restrict__ link, const float* __restrict__ ww,
    const float* __restrict__ prec, const float* __restrict__ rdw,
    float* __restrict__ fwdv)
{
  __shared__ float s_ww[512], s_prec[512], s_rw[4 * 512];
  int b = blockIdx.x, tid = threadIdx.x;
  for (int t = tid; t < 512; t += 256) { s_ww[t] = ww[b * M_ + t]; s_prec[t] = prec[b * M_ + t]; }
  for (int t = tid; t < 2048; t += 256) s_rw[t] = rdw[b * 2048 + t];
  __syncthreads();

  int i    = blockIdx.y * 8 + (tid >> 5);
  int lane = tid & 31;
  float wwi = s_ww[i];
  float a0 = 0.f, a1 = 0.f, a2 = 0.f, a3 = 0.f;
  const float* lrow = link + ((size_t)(b * M_ + i)) * M_;
  for (int j = lane; j < M_; j += 32) {
    float lij = lrow[j];
    float L = (j == i) ? 0.f : ((1.f - wwi - s_ww[j]) * lij + wwi * s_prec[j]);
    a0 += L * s_rw[j];
    a1 += L * s_rw[512 + j];
    a2 += L * s_rw[1024 + j];
    a3 += L * s_rw[1536 + j];
  }
  for (int off = 16; off > 0; off >>= 1) {
    a0 += __shfl_down(a0, off);
    a1 += __shfl_down(a1, off);
    a2 += __shfl_down(a2, off);
    a3 += __shfl_down(a3, off);
  }
  if (lane == 0) {
    fwdv[(b * R_ + 0) * M_ + i] = a0;
    fwdv[(b * R_ + 1) * M_ + i] = a1;
    fwdv[(b * R_ + 2) * M_ + i] = a2;
    fwdv[(b * R_ + 3) * M_ + i] = a3;
  }
}

// ---------------------------------------------------------------------------
// Kernel 9: bwd[b,r,j] = sum_{i!=j} rw[r,i]*L[i,j]. Thread = column j
// (coalesced row sweeps over link).
// ---------------------------------------------------------------------------
__global__ __launch_bounds__(256) void link_bwd_kernel(
    const float* __restrict__ link, const float* __restrict__ ww,
    const float* __restrict__ prec, const float* __restrict__ rdw,
    float* __restrict__ bwdv)
{
  __shared__ float s_ww[512], s_rw[4 * 512];
  int b = blockIdx.x, tid = threadIdx.x;
  for (int t = tid; t < 512; t += 256) s_ww[t] = ww[b * M_ + t];
  for (int t = tid; t < 2048; t += 256) s_rw[t] = rdw[b * 2048 + t];
  __syncthreads();

  int j = blockIdx.y * 256 + tid;
  float wwj = s_ww[j];
  float pj  = prec[b * M_ + j];
  float a0 = 0.f, a1 = 0.f, a2 = 0.f, a3 = 0.f;
  const float* lcol = link + ((size_t)b * M_) * M_ + j;
  for (int i = 0; i < M_; ++i) {
    float lij = lcol[(size_t)i * M_];
    float L = (i == j) ? 0.f : ((1.f - s_ww[i] - wwj) * lij + s_ww[i] * pj);
    a0 += s_rw[i] * L;
    a1 += s_rw[512 + i] * L;
    a2 += s_rw[1024 + i] * L;
    a3 += s_rw[1536 + i] * L;
  }
  bwdv[(b * R_ + 0) * M_ + j] = a0;
  bwdv[(b * R_ + 1) * M_ + j] = a1;
  bwdv[(b * R_ + 2) * M_ + j] = a2;
  bwdv[(b * R_ + 3) * M_ + j] = a3;
}

// ---------------------------------------------------------------------------
// Kernel 10: new_rw = rm0*bwd + rm1*fwd + rm2*cw; out[b,r,w] = new_rw @ mem
// ---------------------------------------------------------------------------
__global__ __launch_bounds__(256) void readout_kernel(
    const float* __restrict__ cw, const float* __restrict__ fwdv,
    const float* __restrict__ bwdv, const float* __restrict__ rm,
    const float* __restrict__ mem, float* __restrict__ out)
{
  __shared__ float s_nrw[R_ * M_];
  int b = blockIdx.x, tid = threadIdx.x;
  for (int t = tid; t < R_ * M_; t += 256) {
    int r = t >> 9, m = t & (M_ - 1);
    int base = b * R_ + r;
    float m0 = rm[base * 3 + 0], m1 = rm[base * 3 + 1], m2 = rm[base * 3 + 2];
    s_nrw[t] = m0 * bwdv[base * M_ + m] + m1 * fwdv[base * M_ + m] + m2 * cw[base * M_ + m];
  }
  __syncthreads();

  int r = tid >> 6, w = tid & (W_ - 1);
  const float* mb = mem + (size_t)b * M_ * W_ + w;
  const float* nr = s_nrw + r * M_;
  float acc = 0.f;
  for (int m = 0; m < M_; ++m) acc = fmaf(nr[m], mb[(size_t)m * W_], acc);
  out[(b * R_ + r) * W_ + w] = acc;
}

// ---------------------------------------------------------------------------
extern "C" void kernel_launch(void* const* d_in, const int* in_sizes, int n_in,
                              void* d_out, int out_size, void* d_ws, size_t ws_size,
                              hipStream_t stream)
{
  (void)in_sizes; (void)n_in; (void)out_size; (void)ws_size;

  const float* x      = (const float*)d_in[0];
  const float* memory = (const float*)d_in[1];
  const float* link   = (const float*)d_in[2];
  const float* prec   = (const float*)d_in[3];
  const float* rdw    = (const float*)d_in[4];
  const float* wwts   = (const float*)d_in[5];
  const float* usage  = (const float*)d_in[6];
  const float* Wp[10]; const float* Bp[10];
  for (int i = 0; i < 10; ++i) { Wp[i] = (const float*)d_in[7 + 2 * i]; Bp[i] = (const float*)d_in[8 + 2 * i]; }

  char* ws = (char*)d_ws;
  size_t off = 0;
  auto take = [&](size_t bytes) -> float* {
    float* p = (float*)(ws + off);
    off = (off + bytes + 255) & ~(size_t)255;
    return p;
  };
  float* wcat = take((size_t)OPAD * IN_ * 4);
  float* bias = take((size_t)OPAD * 4);
  float* proj = take((size_t)B_ * OPAD * 4);
  float* kn   = take((size_t)B_ * R_ * W_ * 4);
  float* wkn  = take((size_t)B_ * W_ * 4);
  float* rs   = take((size_t)B_ * R_ * 4);
  float* wstr = take((size_t)B_ * 4);
  float* er   = take((size_t)B_ * W_ * 4);
  float* wv   = take((size_t)B_ * W_ * 4);
  float* ag   = take((size_t)B_ * 4);
  float* wg   = take((size_t)B_ * 4);
  float* rm   = take((size_t)B_ * R_ * 3 * 4);
  float* u2   = take((size_t)B_ * M_ * 4);
  float* wcw  = take((size_t)B_ * M_ * 4);
  float* ww   = take((size_t)B_ * M_ * 4);
  float* mem  = take((size_t)B_ * M_ * W_ * 4);
  float* cw   = take((size_t)B_ * R_ * M_ * 4);
  float* fwdv = take((size_t)B_ * R_ * M_ * 4);
  float* bwdv = take((size_t)B_ * R_ * M_ * 4);

  pack_w_kernel<<<(OPAD * IN_ + 255) / 256, 256, 0, stream>>>(
      Wp[0], Wp[1], Wp[2], Wp[3], Wp[4], Wp[5], Wp[6], Wp[7], Wp[8], Wp[9],
      Bp[0], Bp[1], Bp[2], Bp[3], Bp[4], Bp[5], Bp[6], Bp[7], Bp[8], Bp[9],
      wcat, bias);
  proj_gemm_wmma<<<dim3(B_ / 16, OPAD / 16), 32, 0, stream>>>(x, wcat, proj);
  act_kernel<<<B_, 256, 0, stream>>>(proj, bias, usage, wwts, rdw,
                                     kn, wkn, rs, wstr, er, wv, ag, wg, rm, u2);
  wcw_kernel<<<B_, 512, 0, stream>>>(memory, wkn, wstr, wcw);
  alloc_kernel<<<B_, 256, 0, stream>>>(u2, wcw, ag, wg, ww);
  memupd_kernel<<<(B_ * M_ * W_) / 256, 256, 0, stream>>>(memory, ww, er, wv, mem);
  cw_kernel<<<B_, 512, 0, stream>>>(mem, kn, rs, cw);
  link_fwd_kernel<<<dim3(B_, M_ / 8), 256, 0, stream>>>(link, ww, prec, rdw, fwdv);
  link_bwd_kernel<<<dim3(B_, M_ / 256), 256, 0, stream>>>(link, ww, prec, rdw, bwdv);
  readout_kernel<<<B_, 256, 0, stream>>>(cw, fwdv, bwdv, rm, mem, (float*)d_out);
}